// MeanAggregator_67757404061979
// MI455X (gfx1250) — compile-verified
//
#include <hip/hip_runtime.h>

// MeanAggregator on MI455X (gfx1250): HBM-roofline gather+reduce (~58 us floor:
// 1.28 GB gathered reads + 51 MB NT writes @ 23.3 TB/s). CDNA5 paths:
//   - GLOBAL_LOAD_ASYNC_TO_LDS_B128 (GVS form: scalar base + lane offset VGPR
//     + imm offset), 25 rows in flight per wave on ASYNCcnt, no dest VGPRs.
//   - s_wait_asynccnt, then ds_load_b128 + packed VALU accumulate.
//   - nontemporal b128 store (write-once output; preserve L2 for the table).

#define NUM_SAMPLE      25
#define FEAT_DIM        128
#define WAVES_PER_BLOCK 4
#define BLOCK_THREADS   (WAVES_PER_BLOCK * 32)

typedef float v4f __attribute__((ext_vector_type(4)));

__device__ __forceinline__ void wait_async_zero() {
#if __has_builtin(__builtin_amdgcn_s_wait_asynccnt)
  __builtin_amdgcn_s_wait_asynccnt(0);
#else
  asm volatile("s_wait_asynccnt 0x0" ::: "memory");
#endif
  asm volatile("" ::: "memory");  // LDS data now visible to this wave
}

// Issue row S: each lane copies 16 bytes of row jb[S] to the LDS address in vdst plus S*512, where
// sbase is backed off by S*512 because the shared immediate is also added to the global address.
// Template parameter guarantees a compile-time immediate for offset:.
template <int S, int N>
struct GatherIssue {
  static __device__ __forceinline__ void run(const int (&jb)[NUM_SAMPLE],
                                             uint64_t feat_base,
                                             uint32_t lds_dst,
                                             uint32_t lane_off) {
    const uint64_t sbase = feat_base +
                           (((uint64_t)(unsigned)jb[S]) << 9) -
                           (uint64_t)(S * 512);
    asm volatile("global_load_async_to_lds_b128 %0, %1, %2 offset:%3"
                 :: "v"(lds_dst), "v"(lane_off), "s"(sbase), "n"(S * 512));
    GatherIssue<S + 1, N>::run(jb, feat_base, lds_dst, lane_off);
  }
};
template <int N>
struct GatherIssue<N, N> {
  static __device__ __forceinline__ void run(const int (&)[NUM_SAMPLE],
                                             uint64_t, uint32_t, uint32_t) {}
};

__global__ __launch_bounds__(BLOCK_THREADS)
void mean_agg_async_kernel(const float* __restrict__ features,
                           const int*   __restrict__ neigh_idx,
                           float*       __restrict__ out,
                           int n_nodes) {
  // Per-wave staging: 25 rows x 512 B. 4 waves -> 51200 B/block (~6 blk/WGP).
  __shared__ v4f smem[WAVES_PER_BLOCK][NUM_SAMPLE][32];

  const int wave = threadIdx.x >> 5;
  const int lane = threadIdx.x & 31;

  // One wave per node; scalarize node id so index fetches become SMEM loads.
  const int node = __builtin_amdgcn_readfirstlane(
      (int)(blockIdx.x * WAVES_PER_BLOCK + wave));
  if (node >= n_nodes) return;

  // Batch ALL 25 uniform index loads before any asm: one SMEM burst,
  // one kmcnt wait (previous version serialized 25 round-trips).
  const int* __restrict__ row_idx = neigh_idx + (size_t)node * NUM_SAMPLE;
  int jb[NUM_SAMPLE];
#pragma unroll
  for (int s = 0; s < NUM_SAMPLE; ++s) jb[s] = row_idx[s];

  // Loop-invariant per-lane VGPRs; all per-row address math is SALU.
  const uint32_t lane_off = (uint32_t)(lane << 4);                 // vaddr
  const uint32_t lds_dst  = (uint32_t)(uintptr_t)&smem[wave][0][lane];

  GatherIssue<0, NUM_SAMPLE>::run(jb, (uint64_t)(uintptr_t)features,
                                  lds_dst, lane_off);

  wait_async_zero();

  // Reduce: 25 x ds_load_b128 + packed adds per lane.
  v4f acc = {0.0f, 0.0f, 0.0f, 0.0f};
#pragma unroll
  for (int s = 0; s < NUM_SAMPLE; ++s) {
    acc += smem[wave][s][lane];
  }
  acc *= (1.0f / (float)NUM_SAMPLE);

  // Write-once output: nontemporal keeps the 192 MB L2 for the feature table.
  v4f* orow = (v4f*)(out + (size_t)node * FEAT_DIM) + lane;
  __builtin_nontemporal_store(acc, orow);
}

extern "C" void kernel_launch(void* const* d_in, const int* in_sizes, int n_in,
                              void* d_out, int out_size, void* d_ws,
                              size_t ws_size, hipStream_t stream) {
  (void)n_in; (void)out_size; (void)d_ws; (void)ws_size;

  const float* features  = (const float*)d_in[0];
  const int*   neigh_idx = (const int*)d_in[1];
  float*       out       = (float*)d_out;

  const int n_nodes = in_sizes[1] / NUM_SAMPLE;   // 100000
  const int grid = (n_nodes + WAVES_PER_BLOCK - 1) / WAVES_PER_BLOCK;

  hipLaunchKernelGGL(mean_agg_async_kernel, dim3(grid), dim3(BLOCK_THREADS), 0,
                     stream, features, neigh_idx, out, n_nodes);
}